// LatticeConstrainedLayer_5592047420119
// MI455X (gfx1250) — compile-verified
//
#include <hip/hip_runtime.h>

// E8 lattice closest-point decode, one 8-dim row per lane.
// Memory-bound streaming kernel: 256 MB total traffic -> ~11 us roofline at
// 23.3 TB/s. Loads keep default temporal hint (input fits MI455X's 192 MB L2
// and is reused across timed replays); stores are non-temporal so the
// write-once output does not evict the input from L2.

typedef float v4f __attribute__((ext_vector_type(4)));

// Closest point of D8 (integer vectors with even coordinate sum).
// Matches the JAX reference exactly:
//   f = round-half-even(x); d = x - f
//   worst = first argmax |d|; step = (d[worst] >= 0) ? +1 : -1
//   if sum(f) odd: flip worst coord by step
__device__ __forceinline__ void closestD8(const float* __restrict__ x,
                                          float* __restrict__ c) {
    float f[8], d[8];
#pragma unroll
    for (int i = 0; i < 8; ++i) {
        f[i] = rintf(x[i]);           // v_rndne_f32 == jnp.round (RNE)
        d[i] = x[i] - f[i];
    }

    // First-occurrence argmax of |d| (strict '>' keeps the first max, like
    // jnp.argmax), tracking the signed residual for the step direction.
    float bestAbs = __builtin_fabsf(d[0]);
    float bestD   = d[0];
    int   bestIdx = 0;
#pragma unroll
    for (int i = 1; i < 8; ++i) {
        float a = __builtin_fabsf(d[i]);
        bool gt = a > bestAbs;
        bestAbs = gt ? a    : bestAbs;
        bestD   = gt ? d[i] : bestD;
        bestIdx = gt ? i    : bestIdx;
    }

    // Parity of the (exactly integer-valued) coordinate sum.
    float s = 0.0f;
#pragma unroll
    for (int i = 0; i < 8; ++i) s += f[i];
    int odd = ((int)s) & 1;           // exact: |s| small, integer-valued

    // d >= 0 -> +1 (true for -0.0 as well, matching jnp.where(d >= 0,...))
    float step = (bestD >= 0.0f) ? 1.0f : -1.0f;
    int fix = odd ? bestIdx : 8;      // 8 == "no coordinate fixed"
#pragma unroll
    for (int i = 0; i < 8; ++i)
        c[i] = (i == fix) ? (f[i] + step) : f[i];
}

__global__ __launch_bounds__(256) void e8_closest_kernel(
    const float* __restrict__ in, float* __restrict__ out, int nrows) {
    int r = blockIdx.x * blockDim.x + threadIdx.x;
    if (r >= nrows) return;

    // One row (32 B) per lane: two b128 loads, contiguous 1 KB per wave32.
    const v4f* ip = (const v4f*)in + (size_t)2 * (size_t)r;
    v4f a = ip[0];
    v4f b = ip[1];

    float x[8], xs[8], c0[8], c1[8];
#pragma unroll
    for (int i = 0; i < 4; ++i) { x[i] = a[i]; x[i + 4] = b[i]; }
#pragma unroll
    for (int i = 0; i < 8; ++i) xs[i] = x[i] - 0.5f;

    closestD8(x,  c0);                // D8 coset
    closestD8(xs, c1);                // D8 + (1/2,...,1/2) coset
#pragma unroll
    for (int i = 0; i < 8; ++i) c1[i] += 0.5f;

    float d0 = 0.0f, d1 = 0.0f;
#pragma unroll
    for (int i = 0; i < 8; ++i) {
        float e0 = x[i] - c0[i]; d0 = __builtin_fmaf(e0, e0, d0);
        float e1 = x[i] - c1[i]; d1 = __builtin_fmaf(e1, e1, d1);
    }
    bool pick0 = d0 <= d1;            // tie -> coset 0, like the reference

    v4f o0, o1;
#pragma unroll
    for (int i = 0; i < 4; ++i) {
        o0[i] = pick0 ? c0[i]     : c1[i];
        o1[i] = pick0 ? c0[i + 4] : c1[i + 4];
    }

    // Non-temporal b128 stores: write-once output must not evict the
    // (L2-resident, 128 MB < 192 MB) input across timed replays.
    v4f* op = (v4f*)out + (size_t)2 * (size_t)r;
    __builtin_nontemporal_store(o0, op);
    __builtin_nontemporal_store(o1, op + 1);
}

extern "C" void kernel_launch(void* const* d_in, const int* in_sizes, int n_in,
                              void* d_out, int out_size, void* d_ws, size_t ws_size,
                              hipStream_t stream) {
    const float* x = (const float*)d_in[0];
    float* out = (float*)d_out;
    int nrows = in_sizes[0] / 8;                 // 4,000,000 rows of dim 8
    const int block = 256;                        // 8 wave32s per block
    int grid = (nrows + block - 1) / block;
    e8_closest_kernel<<<grid, block, 0, stream>>>(x, out, nrows);
}